// SinkhornOT_49211735277882
// MI455X (gfx1250) — compile-verified
//
#include <hip/hip_runtime.h>
#include <hip/hip_fp16.h>

#define BB 8
#define NN 2048
#define DD 64
#define REGC 0.1f
#define ITERS 50

typedef float v2f __attribute__((ext_vector_type(2)));
typedef float v8f __attribute__((ext_vector_type(8)));

// ---------------------------------------------------------------------------
// Init: per-batch max = 0, u = v = 1/N   (must re-run every call: harness does
// not re-poison, and we must be deterministic)
// ---------------------------------------------------------------------------
__global__ void sk_init_kernel(int* __restrict__ maxC, float* __restrict__ u,
                               float* __restrict__ v) {
    int i = blockIdx.x * blockDim.x + threadIdx.x;
    if (i < BB) maxC[i] = 0;
    if (i < BB * NN) {
        u[i] = 1.0f / NN;
        v[i] = 1.0f / NN;
    }
}

// ---------------------------------------------------------------------------
// Row squared-norms of x0 and x1
// ---------------------------------------------------------------------------
__global__ void sk_rowsq_kernel(const float* __restrict__ x0,
                                const float* __restrict__ x1,
                                float* __restrict__ sq0, float* __restrict__ sq1) {
    int i = blockIdx.x * blockDim.x + threadIdx.x;
    if (i >= BB * NN) return;
    const float* p0 = x0 + (size_t)i * DD;
    const float* p1 = x1 + (size_t)i * DD;
    float s0 = 0.0f, s1 = 0.0f;
#pragma unroll
    for (int k = 0; k < DD; ++k) {
        float a = p0[k]; s0 += a * a;
        float b = p1[k]; s1 += b * b;
    }
    sq0[i] = s0;
    sq1[i] = s1;
}

// ---------------------------------------------------------------------------
// Cost tiles via V_WMMA_F32_16X16X4_F32.
// One wave computes one 16x16 tile of inner = x0 . x1^T (K=64 => 16 WMMAs),
// then C = sqrt(max(sq0_i + sq1_j - 2*inner, 0)), stores fp16 C, and
// atomicMax's the per-batch max (non-negative floats => int-bit compare OK).
//
// f32 A 16x4 layout: lanes 0-15 hold K=0,1 ; lanes 16-31 hold K=2,3 (2 VGPRs)
// => each lane loads a float2 from its row at k-offset (4t + 2*half).
// B 4x16 mirrors this with x1 rows (column tile).
// ---------------------------------------------------------------------------
__global__ void sk_cost_tile_kernel(const float* __restrict__ x0,
                                    const float* __restrict__ x1,
                                    const float* __restrict__ sq0,
                                    const float* __restrict__ sq1,
                                    __half* __restrict__ Kbuf,
                                    int* __restrict__ maxC) {
    const int TPB = (NN / 16) * (NN / 16);              // tiles per batch
    int wave = threadIdx.x >> 5;
    int lane = threadIdx.x & 31;
    int gt   = blockIdx.x * 8 + wave;                   // global tile id
    int b    = gt / TPB;
    int t    = gt % TPB;
    int row0 = (t / (NN / 16)) * 16;
    int col0 = (t % (NN / 16)) * 16;
    int hh   = lane >> 4;                               // half-wave select
    int lrow = lane & 15;

    const float* ap = x0 + ((size_t)b * NN + row0 + lrow) * DD + 2 * hh;
    const float* bp = x1 + ((size_t)b * NN + col0 + lrow) * DD + 2 * hh;

    v8f acc = {};
#pragma unroll
    for (int k = 0; k < DD; k += 4) {
        v2f av = *(const v2f*)(ap + k);
        v2f bv = *(const v2f*)(bp + k);
        acc = __builtin_amdgcn_wmma_f32_16x16x4_f32(
            /*neg_a=*/false, av, /*neg_b=*/false, bv,
            /*c_mod=*/(short)0, acc, /*reuse_a=*/false, /*reuse_b=*/false);
    }

    // C/D f32 layout: VGPR r, lanes 0-15 -> M=r, lanes 16-31 -> M=r+8, N=lane%16
    int   col  = col0 + lrow;
    float sq1c = sq1[b * NN + col];
    float mx   = 0.0f;
#pragma unroll
    for (int r = 0; r < 8; ++r) {
        int   row = row0 + r + hh * 8;
        float d2  = sq0[b * NN + row] + sq1c - 2.0f * acc[r];
        float cv  = sqrtf(fmaxf(d2, 0.0f));
        Kbuf[((size_t)b * NN + row) * NN + col] = __float2half(cv);
        mx = fmaxf(mx, cv);
    }
    for (int m = 16; m > 0; m >>= 1)
        mx = fmaxf(mx, __shfl_xor(mx, m, 32));
    if (lane == 0) atomicMax(&maxC[b], __float_as_int(mx));
}

// ---------------------------------------------------------------------------
// K = exp(-C / (REG * (maxC + 1e-8))) in place, plus LDS-tiled transpose into
// KT so both Sinkhorn sweep directions are row-major.
// Block = 32x8 threads handles one 32x32 tile.
// ---------------------------------------------------------------------------
__global__ void sk_exp_transpose_kernel(__half* __restrict__ Kbuf,
                                        __half* __restrict__ KTbuf,
                                        const int* __restrict__ maxC) {
    __shared__ float sm[32][33];
    const int TPB = (NN / 32) * (NN / 32);
    int b  = blockIdx.x / TPB;
    int t  = blockIdx.x % TPB;
    int ti = t / (NN / 32);
    int tj = t % (NN / 32);
    float scale = 1.0f / (REGC * (__int_as_float(maxC[b]) + 1e-8f));
    int tx = threadIdx.x, ty = threadIdx.y;
    const size_t base = (size_t)b * NN * NN;
#pragma unroll
    for (int s = 0; s < 4; ++s) {
        int rl  = ty + s * 8;
        size_t idx = base + (size_t)(ti * 32 + rl) * NN + (tj * 32 + tx);
        float c = __half2float(Kbuf[idx]);
        float k = __expf(-c * scale);
        Kbuf[idx] = __float2half(k);
        sm[rl][tx] = k;
    }
    __syncthreads();
#pragma unroll
    for (int s = 0; s < 4; ++s) {
        int rl = ty + s * 8;
        // KT[b][tj*32+rl][ti*32+tx] = K[b][ti*32+tx][tj*32+rl]
        KTbuf[base + (size_t)(tj * 32 + rl) * NN + (ti * 32 + tx)] =
            __float2half(sm[tx][rl]);
    }
}

// ---------------------------------------------------------------------------
// One Sinkhorn half-step: vout[b][row] = 1 / (sum_j M[b][row][j]*vin[b][j] + eps)
// Block = 8 waves, 8 consecutive rows of one batch; vin staged in LDS (8 KB).
// ---------------------------------------------------------------------------
__global__ void sk_gemv_recip_kernel(const __half* __restrict__ M,
                                     const float* __restrict__ vin,
                                     float* __restrict__ vout) {
    __shared__ float ldsv[NN];
    int b    = blockIdx.x >> 8;          // 256 row-blocks per batch
    int rblk = blockIdx.x & 255;
    for (int i = threadIdx.x; i < NN; i += 256) ldsv[i] = vin[b * NN + i];
    __syncthreads();
    int wave = threadIdx.x >> 5, lane = threadIdx.x & 31;
    int row  = rblk * 8 + wave;
    const __half2* Mrow = (const __half2*)(M + ((size_t)b * NN + row) * NN);
    float sum = 0.0f;
#pragma unroll 4
    for (int t = 0; t < NN / 64; ++t) {
        int    j2 = lane + 32 * t;
        float2 kf = __half22float2(Mrow[j2]);
        sum += kf.x * ldsv[2 * j2] + kf.y * ldsv[2 * j2 + 1];
    }
    for (int m = 16; m > 0; m >>= 1) sum += __shfl_xor(sum, m, 32);
    if (lane == 0) vout[b * NN + row] = 1.0f / (sum + 1e-10f);
}

// ---------------------------------------------------------------------------
// matchings[b][i] = argmax_j u_i*K_ij*v_j == argmax_j K_ij*v_j  (u_i > 0).
// First-occurrence tie-break like jnp.argmax. Output written as float.
// ---------------------------------------------------------------------------
__global__ void sk_argmax_kernel(const __half* __restrict__ K,
                                 const float* __restrict__ v,
                                 float* __restrict__ out) {
    __shared__ float ldsv[NN];
    int b    = blockIdx.x >> 8;
    int rblk = blockIdx.x & 255;
    for (int i = threadIdx.x; i < NN; i += 256) ldsv[i] = v[b * NN + i];
    __syncthreads();
    int wave = threadIdx.x >> 5, lane = threadIdx.x & 31;
    int row  = rblk * 8 + wave;
    const __half2* Krow = (const __half2*)(K + ((size_t)b * NN + row) * NN);
    float best = -1.0f;
    int   bidx = 0x7fffffff;
    for (int t = 0; t < NN / 64; ++t) {
        int    j2 = lane + 32 * t;
        float2 kf = __half22float2(Krow[j2]);
        float  p0 = kf.x * ldsv[2 * j2];
        float  p1 = kf.y * ldsv[2 * j2 + 1];
        if (p0 > best || (p0 == best && 2 * j2 < bidx))     { best = p0; bidx = 2 * j2; }
        if (p1 > best || (p1 == best && 2 * j2 + 1 < bidx)) { best = p1; bidx = 2 * j2 + 1; }
    }
    for (int m = 16; m > 0; m >>= 1) {
        float ob = __shfl_xor(best, m, 32);
        int   oi = __shfl_xor(bidx, m, 32);
        if (ob > best || (ob == best && oi < bidx)) { best = ob; bidx = oi; }
    }
    if (lane == 0) out[b * NN + row] = (float)bidx;
}

// ---------------------------------------------------------------------------
extern "C" void kernel_launch(void* const* d_in, const int* in_sizes, int n_in,
                              void* d_out, int out_size, void* d_ws, size_t ws_size,
                              hipStream_t stream) {
    const float* x0 = (const float*)d_in[0];
    const float* x1 = (const float*)d_in[1];

    char* ws = (char*)d_ws;
    const size_t KBYTES = (size_t)BB * NN * NN * sizeof(__half); // 67 MB
    __half* Kbuf  = (__half*)ws;
    __half* KTbuf = (__half*)(ws + KBYTES);
    float*  sq0   = (float*)(ws + 2 * KBYTES);
    float*  sq1   = sq0 + BB * NN;
    float*  u     = sq1 + BB * NN;
    float*  v     = u + BB * NN;
    int*    maxC  = (int*)(v + BB * NN);

    sk_init_kernel<<<64, 256, 0, stream>>>(maxC, u, v);
    sk_rowsq_kernel<<<64, 256, 0, stream>>>(x0, x1, sq0, sq1);

    // 8 batches * 128*128 tiles / 8 waves per block
    sk_cost_tile_kernel<<<BB * 128 * 128 / 8, 256, 0, stream>>>(
        x0, x1, sq0, sq1, Kbuf, maxC);

    sk_exp_transpose_kernel<<<BB * 64 * 64, dim3(32, 8), 0, stream>>>(
        Kbuf, KTbuf, maxC);

    for (int it = 0; it < ITERS; ++it) {
        sk_gemv_recip_kernel<<<BB * 256, 256, 0, stream>>>(Kbuf, v, u);
        sk_gemv_recip_kernel<<<BB * 256, 256, 0, stream>>>(KTbuf, u, v);
    }

    sk_argmax_kernel<<<BB * 256, 256, 0, stream>>>(Kbuf, v, (float*)d_out);
}